// FinePreprocess_37263136260362
// MI455X (gfx1250) — compile-verified
//
#include <hip/hip_runtime.h>
#include <hip/hip_bf16.h>
#include <stdint.h>

// Problem constants (from reference)
#define CROPSZ   8
#define NVIEW    16
#define CCH      32
#define HH       512
#define WW       512
#define NTRACK   2048
#define NCROPS   (NVIEW * NTRACK)      // 32768
#define WIN      10                    // 10x10 window covers all bilinear taps
#define WELEM    (CCH * WIN * WIN)     // 3200 floats per window
#define CROPS_PER_BLOCK 8

typedef unsigned int v4u __attribute__((ext_vector_type(4)));
typedef int          v8i __attribute__((ext_vector_type(8)));
typedef int          v4i __attribute__((ext_vector_type(4)));

// Issue one TDM tensor load: 3-D tile (x=10, y=10, c=32) of f32 from the
// selected feature image into LDS at byte offset lds_off.
// Descriptor layout per CDNA5 ISA.
__device__ __forceinline__ void tdm_issue(const float* __restrict__ feats,
                                          const float* __restrict__ pts,
                                          const int*   __restrict__ idxs,
                                          int n, unsigned lds_off) {
    int   b  = idxs[n];
    float px = pts[2 * n + 0];
    float py = pts[2 * n + 1];
    int xb = (int)floorf(px - 4.0f);   // window origin (never OOB: pts in [8,504])
    int yb = (int)floorf(py - 4.0f);
    b  = __builtin_amdgcn_readfirstlane(b);
    xb = __builtin_amdgcn_readfirstlane(xb);
    yb = __builtin_amdgcn_readfirstlane(yb);

    unsigned long long ga = (unsigned long long)(uintptr_t)feats
        + 4ull * ((unsigned long long)b * (CCH * HH * WW)
                  + (unsigned long long)yb * WW
                  + (unsigned long long)xb);

    // ---- D# group 0 (4 dwords) ----
    v4u g0;
    g0[0] = 1u;                                   // count=1, no gather/iterate
    g0[1] = lds_off;                              // LDS byte address
    g0[2] = (unsigned)ga;                         // global_addr[31:0]
    g0[3] = (unsigned)((ga >> 32) & 0x01FFFFFFu)  // global_addr[56:32]
          | 0x80000000u;                          // type = 2 ("image")

    // ---- D# group 1 (8 dwords) ----
    v8i g1;
    g1[0] = 0x00020000;                 // wg_mask=0, data_size=2 (4 bytes)
    g1[1] = (int)(WW << 16);            // tensor_dim0[15:0] = 512
    g1[2] = (int)(HH << 16);            // tensor_dim0[31:16]=0 | tensor_dim1[15:0]=512
    g1[3] = (int)(WIN << 16);           // tensor_dim1[31:16]=0 | tile_dim0=10
    g1[4] = (int)(WIN | (CCH << 16));   // tile_dim1=10 | tile_dim2=32
    g1[5] = WW;                         // tensor_dim0_stride[31:0] = 512 (row)
    g1[6] = 0;                          // stride0[47:32]=0 | stride1[15:0]=0
    g1[7] = (int)((HH * WW) >> 16);     // tensor_dim1_stride[47:16] (=512*512, channel)

    // ---- D# group 2/3 (3-D tile: tensor_dim2 = C) ----
    v4i g2 = { CCH, 0, 0, 0 };          // tensor_dim2=32; tile_dim3=0 (unused)
    v4i g3 = { 0, 0, 0, 0 };

#if defined(__HIP_DEVICE_COMPILE__)
    // Unconditional in the device pass: a missing builtin must fail LOUDLY here.
# if defined(__clang_major__) && (__clang_major__ >= 23)
    v8i g4 = { 0, 0, 0, 0, 0, 0, 0, 0 };
    __builtin_amdgcn_tensor_load_to_lds(g0, g1, g2, g3, g4, 0);
# else
    __builtin_amdgcn_tensor_load_to_lds(g0, g1, g2, g3, 0);
# endif
#else
    (void)g0; (void)g1; (void)g2; (void)g3;   // host pass: never executed
#endif
}

__device__ __forceinline__ void wait_tensorcnt0() {
#if defined(__HIP_DEVICE_COMPILE__)
    __builtin_amdgcn_s_wait_tensorcnt(0);
#endif
}
__device__ __forceinline__ void wait_tensorcnt1() {
#if defined(__HIP_DEVICE_COMPILE__)
    __builtin_amdgcn_s_wait_tensorcnt(1);
#endif
}

__global__ __launch_bounds__(256, 1)
void FinePreprocess_bilinear_crops_kernel(const float* __restrict__ feats,
                                          const float* __restrict__ pts,
                                          const int*   __restrict__ idxs,
                                          float*       __restrict__ out) {
    __shared__ float smem[2][WELEM];           // double-buffered windows (25.6 KB)

    const int tid  = threadIdx.x;
    const int lane = tid & 31;                 // = channel (wave32: 32 lanes = 32 ch)
    const int wv   = tid >> 5;                 // wave id = crop row j (8 waves)
    const int base = blockIdx.x * CROPS_PER_BLOCK;

    const unsigned lds0 = (unsigned)(uintptr_t)&smem[0][0];
    const unsigned lds1 = (unsigned)(uintptr_t)&smem[1][0];

    // Wave-uniform scalar branch (TDM ignores EXEC -> must be a real branch).
    const int swv = __builtin_amdgcn_readfirstlane(wv);

    if (swv == 0) {
        tdm_issue(feats, pts, idxs, base, lds0);
    }

    for (int k = 0; k < CROPS_PER_BLOCK; ++k) {
        const int n = base + k;
        const float* __restrict__ buf = smem[k & 1];

        if (swv == 0) {
            if (k + 1 < CROPS_PER_BLOCK) {
                tdm_issue(feats, pts, idxs, n + 1, (k & 1) ? lds0 : lds1);
                wait_tensorcnt1();                 // crop k's DMA done
            } else {
                wait_tensorcnt0();
            }
        }
        __syncthreads();   // window k visible to all waves

        // Per-crop uniform interpolation setup
        float px = pts[2 * n + 0];
        float py = pts[2 * n + 1];
        int   xb = (int)floorf(px - 4.0f);
        int   yb = (int)floorf(py - 4.0f);

        float ysv = py + (-4.0f + (float)wv * (8.0f / 7.0f));
        float y0  = floorf(ysv);
        float fy  = ysv - y0;
        int   yo  = (int)y0 - yb;              // in [0, 8]

        const float* __restrict__ rowp = buf + lane * (WIN * WIN) + yo * WIN;
        float* __restrict__ op = out + (((size_t)n * 64 + (size_t)wv * 8) * 32 + lane);

#pragma unroll
        for (int i = 0; i < 8; ++i) {
            float xsv = px + (-4.0f + (float)i * (8.0f / 7.0f));
            float x0  = floorf(xsv);
            float fx  = xsv - x0;
            int   xo  = (int)x0 - xb;          // in [0, 8]

            float v00 = rowp[xo];
            float v01 = rowp[xo + 1];
            float v10 = rowp[xo + WIN];
            float v11 = rowp[xo + WIN + 1];

            float top = v00 * (1.0f - fx) + v01 * fx;
            float bot = v10 * (1.0f - fx) + v11 * fx;
            float val = top * (1.0f - fy) + bot * fy;

            // streaming output: bypass caches (keep L2 for feature gathers)
            __builtin_nontemporal_store(val, op + (size_t)i * 32);
        }
        __syncthreads();   // everyone done with buf before it is re-filled
    }
}

extern "C" void kernel_launch(void* const* d_in, const int* in_sizes, int n_in,
                              void* d_out, int out_size, void* d_ws, size_t ws_size,
                              hipStream_t stream) {
    (void)in_sizes; (void)n_in; (void)out_size; (void)d_ws; (void)ws_size;
    const float* feats = (const float*)d_in[0];   // (1,16,32,512,512) f32
    const float* pts   = (const float*)d_in[1];   // (1,16,2048,2)     f32
    const int*   idxs  = (const int*)  d_in[2];   // (1,16,2048)       int
    float*       out   = (float*)d_out;           // (1,16,2048,64,32) f32

    dim3 grid(NCROPS / CROPS_PER_BLOCK);          // 4096 blocks
    dim3 block(256);                              // 8 waves of 32
    FinePreprocess_bilinear_crops_kernel<<<grid, block, 0, stream>>>(feats, pts, idxs, out);
}